// GGNN_23210003267730
// MI455X (gfx1250) — compile-verified
//
#include <hip/hip_runtime.h>

#define D 32
#define K3 96          // 3*D
#define ETYPES 8
#define STEPS 5

typedef __attribute__((ext_vector_type(2))) float v2f;
typedef __attribute__((ext_vector_type(8))) float v8f;

__device__ __forceinline__ float sigmoid_f(float x) {
    return 1.0f / (1.0f + __expf(-x));
}

// --- CDNA5 async global->LDS copies (ASYNCcnt-tracked, no VGPR round trip) ---
__device__ __forceinline__ void async_copy_b128(unsigned lds_byte, const float* gbase,
                                                unsigned voff_byte) {
    asm volatile("global_load_async_to_lds_b128 %0, %1, %2"
                 :: "v"(lds_byte), "v"(voff_byte), "s"(gbase) : "memory");
}
__device__ __forceinline__ void async_copy_b64(unsigned lds_byte, const float* gbase,
                                               unsigned voff_byte) {
    asm volatile("global_load_async_to_lds_b64 %0, %1, %2"
                 :: "v"(lds_byte), "v"(voff_byte), "s"(gbase) : "memory");
}
__device__ __forceinline__ void wait_async0() {
    asm volatile("s_wait_asynccnt 0x0" ::: "memory");
}

// ---------------------------------------------------------------------------
// prop = emb[token]   (one float4 per thread)
// ---------------------------------------------------------------------------
__global__ __launch_bounds__(256)
void ggnn_init(const int* __restrict__ token, const float* __restrict__ emb,
               float* __restrict__ prop, int n) {
    int i = blockIdx.x * 256 + threadIdx.x;
    int total = n * (D / 4);
    if (i >= total) return;
    int node = i >> 3;            // D/4 == 8
    int c = i & 7;
    ((float4*)(prop + (size_t)node * D))[c] =
        ((const float4*)(emb + (size_t)token[node] * D))[c];
}

// ---------------------------------------------------------------------------
// zero in_fea / out_fea (contiguous in workspace)
// ---------------------------------------------------------------------------
__global__ __launch_bounds__(256)
void ggnn_zero(float4* __restrict__ p, int total4) {
    int i = blockIdx.x * 256 + threadIdx.x;
    if (i < total4) p[i] = make_float4(0.f, 0.f, 0.f, 0.f);
}

// ---------------------------------------------------------------------------
// Edge messages: fea = h_src @ W_edge[etype]; atomically scattered into
// in_fea (by dst, k=0) and out_fea (by src, k=1).
// W table staged with async b64 copies into LDS layout [i*32+j][t][k]:
//  - lanes with different etype t hit disjoint banks (conflict-free),
//  - the (k=0,k=1) pair is one aligned ds_load_b64.
// ---------------------------------------------------------------------------
__global__ __launch_bounds__(256)
void ggnn_edge(const float* __restrict__ prop,
               const int* __restrict__ src, const int* __restrict__ dst,
               const int* __restrict__ etype, const float* __restrict__ W_edge,
               float* __restrict__ in_fea, float* __restrict__ out_fea, int E) {
    __shared__ __align__(16) float sW[1024 * ETYPES * 2];   // 64 KB: [ij][t][k]
    const unsigned sWb = (unsigned)(uintptr_t)sW;
    // global float2 chunk idx2 = t*1024 + ij  ->  LDS byte (ij*16 + t*2)*4
    for (int idx2 = threadIdx.x; idx2 < 1024 * ETYPES; idx2 += 256) {
        int t  = idx2 >> 10;
        int ij = idx2 & 1023;
        async_copy_b64(sWb + (unsigned)(ij * 64 + t * 8), W_edge, (unsigned)idx2 * 8u);
    }
    wait_async0();
    __syncthreads();

    int e = blockIdx.x * 256 + threadIdx.x;
    if (e >= E) return;
    int s = src[e], d = dst[e], t = etype[e];

    float h[D];
    const float4* hp = (const float4*)(prop + (size_t)s * D);
#pragma unroll
    for (int i = 0; i < D / 4; ++i) {
        float4 v = hp[i];
        h[4 * i + 0] = v.x; h[4 * i + 1] = v.y;
        h[4 * i + 2] = v.z; h[4 * i + 3] = v.w;
    }

    float* inb  = in_fea  + (size_t)d * D;
    float* outb = out_fea + (size_t)s * D;
    __builtin_prefetch(inb, 1, 1);     // global_prefetch_b8
    __builtin_prefetch(outb, 1, 1);

    const float* Wt0 = sW + t * 2;     // + ij*16 selects the (i,j) pair
#pragma unroll 4
    for (int j = 0; j < D; ++j) {
        float a0 = 0.f, a1 = 0.f;
#pragma unroll
        for (int i = 0; i < D; ++i) {
            float2 w = *(const float2*)(Wt0 + (i * D + j) * 16);
            a0 += h[i] * w.x;
            a1 += h[i] * w.y;
        }
        unsafeAtomicAdd(inb + j, a0);    // global_atomic_add_f32 (no return)
        unsafeAtomicAdd(outb + j, a1);
    }
}

// ---------------------------------------------------------------------------
// GRU update with fp32 WMMA (v_wmma_f32_16x16x4_f32).
// Block = 128 threads = 4 waves; each wave owns 16 nodes.
// cat = [in_fea | out_fea | prop] (16 x 96) staged by async b128 copies into a
// wave-private LDS tile. GRU weights live in LDS K-pair-interleaved
// ([k/2][col][2]) so every B fragment is one aligned ds_load_b64.
// r,z share the A fragment; A frags for cols 0..63 are cached in VGPRs and
// reused by the t matmul after cat[:,64:96] <- r*prop (wave-private LDS, DS
// ops in-order within a wave -> no barrier).
// ---------------------------------------------------------------------------
__global__ __launch_bounds__(128)
void ggnn_gru_wmma(float* __restrict__ prop,
                   const float* __restrict__ in_fea,
                   const float* __restrict__ out_fea,
                   const float* __restrict__ Wr, const float* __restrict__ br,
                   const float* __restrict__ Wz, const float* __restrict__ bz,
                   const float* __restrict__ Wt, const float* __restrict__ bt,
                   int n) {
    __shared__ __align__(16) float sW[3 * K3 * D];   // interleaved [mat][k/2][col][2]
    __shared__ __align__(16) float sB[3 * D];        // br|bz|bt
    __shared__ __align__(16) float sCat[4][16][K3];  // per-wave cat tiles

    const int tid  = threadIdx.x;
    const int wave = tid >> 5;
    const int lane = tid & 31;

    // weights -> LDS with K-pair interleave: element (k,col) at (k>>1)*64 + col*2 + (k&1)
    {
        const float* gsrc[3] = { Wr, Wz, Wt };
#pragma unroll
        for (int m = 0; m < 3; ++m) {
            const float2* g2 = (const float2*)gsrc[m];
            float* dstm = sW + m * K3 * D;
            for (int i = tid; i < K3 * D / 2; i += 128) {   // 1536 float2
                int k  = i >> 4;            // 16 float2 per row of 32 cols
                int c2 = i & 15;            // cols 2*c2, 2*c2+1
                float2 v = g2[i];
                float* p = dstm + (k >> 1) * (2 * D) + (k & 1);
                p[(2 * c2) * 2]     = v.x;
                p[(2 * c2 + 1) * 2] = v.y;
            }
        }
        if (tid < 3 * D)
            sB[tid] = (tid < D) ? br[tid] : (tid < 2 * D ? bz[tid - D] : bt[tid - 2 * D]);
    }

    // cat tile -> LDS via async b128: 16 rows x 24 chunks of 16B per wave
    const int rowBase = blockIdx.x * 64 + wave * 16;
    const unsigned catB = (unsigned)(uintptr_t)&sCat[wave][0][0];
    for (int i = lane; i < 16 * (K3 / 4); i += 32) {
        int row = i / 24;
        int c4  = i - row * 24;
        int node = rowBase + row;
        unsigned lds = catB + (unsigned)(row * K3 + c4 * 4) * 4u;
        if (node < n) {
            unsigned nb = (unsigned)node * (D * 4u);
            if (c4 < 8)       async_copy_b128(lds, in_fea,  nb + (unsigned)c4 * 16u);
            else if (c4 < 16) async_copy_b128(lds, out_fea, nb + (unsigned)(c4 - 8) * 16u);
            else              async_copy_b128(lds, prop,    nb + (unsigned)(c4 - 16) * 16u);
        } else {
            ((float4*)&sCat[wave][row][0])[c4] = make_float4(0.f, 0.f, 0.f, 0.f);
        }
    }
    wait_async0();
    __syncthreads();

    const int arow  = lane & 15;
    const int ahalf = lane >> 4;                 // lanes 0-15 : K0/K1, 16-31 : K2/K3
    const int bcol  = lane & 15;
    const v2f* catRow2 = (const v2f*)&sCat[wave][arow][0];
    const float* sWr = sW;
    const float* sWz = sW + K3 * D;
    const float* sWt = sW + 2 * K3 * D;

    v2f aC[16];                                  // cached A frags for cols 0..63
    v8f accR0 = {}, accR1 = {}, accZ0 = {}, accZ1 = {};
#pragma unroll
    for (int k4 = 0; k4 < K3 / 4; ++k4) {
        const int p = 2 * k4 + ahalf;            // K-pair index (kk = 2p)
        v2f a = catRow2[p];
        if (k4 < 16) aC[k4] = a;
        v2f b;
        b = ((const v2f*)(sWr + p * (2 * D)))[bcol];
        accR0 = __builtin_amdgcn_wmma_f32_16x16x4_f32(false, a, false, b, (short)0, accR0, false, false);
        b = ((const v2f*)(sWr + p * (2 * D) + 32))[bcol];
        accR1 = __builtin_amdgcn_wmma_f32_16x16x4_f32(false, a, false, b, (short)0, accR1, false, false);
        b = ((const v2f*)(sWz + p * (2 * D)))[bcol];
        accZ0 = __builtin_amdgcn_wmma_f32_16x16x4_f32(false, a, false, b, (short)0, accZ0, false, false);
        b = ((const v2f*)(sWz + p * (2 * D) + 32))[bcol];
        accZ1 = __builtin_amdgcn_wmma_f32_16x16x4_f32(false, a, false, b, (short)0, accZ1, false, false);
    }

    // gates + write r*prop into cat third chunk (element (M,N): M=j+8*ahalf, N=bcol)
    float r0[8], r1[8], z0[8], z1[8], p0[8], p1[8];
#pragma unroll
    for (int j = 0; j < 8; ++j) {
        int m = j + 8 * ahalf;
        float pa = sCat[wave][m][2 * D + bcol];
        float pb = sCat[wave][m][2 * D + 16 + bcol];
        r0[j] = sigmoid_f(accR0[j] + sB[bcol]);
        r1[j] = sigmoid_f(accR1[j] + sB[16 + bcol]);
        z0[j] = sigmoid_f(accZ0[j] + sB[D + bcol]);
        z1[j] = sigmoid_f(accZ1[j] + sB[D + 16 + bcol]);
        p0[j] = pa; p1[j] = pb;
        sCat[wave][m][2 * D + bcol]      = r0[j] * pa;
        sCat[wave][m][2 * D + 16 + bcol] = r1[j] * pb;
    }

    v8f accT0 = {}, accT1 = {};
#pragma unroll
    for (int k4 = 0; k4 < K3 / 4; ++k4) {
        const int p = 2 * k4 + ahalf;
        v2f a = (k4 < 16) ? aC[k4] : catRow2[p];
        v2f b;
        b = ((const v2f*)(sWt + p * (2 * D)))[bcol];
        accT0 = __builtin_amdgcn_wmma_f32_16x16x4_f32(false, a, false, b, (short)0, accT0, false, false);
        b = ((const v2f*)(sWt + p * (2 * D) + 32))[bcol];
        accT1 = __builtin_amdgcn_wmma_f32_16x16x4_f32(false, a, false, b, (short)0, accT1, false, false);
    }

#pragma unroll
    for (int j = 0; j < 8; ++j) {
        int m = j + 8 * ahalf;
        int node = rowBase + m;
        float h0 = tanhf(accT0[j] + sB[2 * D + bcol]);
        float h1 = tanhf(accT1[j] + sB[2 * D + 16 + bcol]);
        float o0 = (1.f - z0[j]) * p0[j] + z0[j] * h0;
        float o1 = (1.f - z1[j]) * p1[j] + z1[j] * h1;
        if (node < n) {
            prop[(size_t)node * D + bcol]      = o0;
            prop[(size_t)node * D + 16 + bcol] = o1;
        }
    }
}

// ---------------------------------------------------------------------------
extern "C" void kernel_launch(void* const* d_in, const int* in_sizes, int n_in,
                              void* d_out, int out_size, void* d_ws, size_t ws_size,
                              hipStream_t stream) {
    const int*   token  = (const int*)d_in[0];
    const int*   etype  = (const int*)d_in[1];
    const int*   src    = (const int*)d_in[2];
    const int*   dst    = (const int*)d_in[3];
    const float* emb    = (const float*)d_in[4];
    const float* W_edge = (const float*)d_in[5];
    const float* W_r    = (const float*)d_in[6];
    const float* b_r    = (const float*)d_in[7];
    const float* W_z    = (const float*)d_in[8];
    const float* b_z    = (const float*)d_in[9];
    const float* W_t    = (const float*)d_in[10];
    const float* b_t    = (const float*)d_in[11];

    const int n = in_sizes[0];   // N_NODES
    const int E = in_sizes[1];   // N_EDGES

    float* prop    = (float*)d_out;                 // [n, D], updated in place
    float* in_fea  = (float*)d_ws;                  // [n, D]
    float* out_fea = in_fea + (size_t)n * D;        // [n, D] (contiguous)

    ggnn_init<<<(n * (D / 4) + 255) / 256, 256, 0, stream>>>(token, emb, prop, n);

    const int total4   = 2 * n * D / 4;             // both fea buffers, float4 units
    const int zeroGrid = (total4 + 255) / 256;
    const int edgeGrid = (E + 255) / 256;
    const int gruGrid  = (n + 63) / 64;

    for (int step = 0; step < STEPS; ++step) {
        ggnn_zero<<<zeroGrid, 256, 0, stream>>>((float4*)in_fea, total4);
        ggnn_edge<<<edgeGrid, 256, 0, stream>>>(prop, src, dst, etype, W_edge,
                                                in_fea, out_fea, E);
        ggnn_gru_wmma<<<gruGrid, 128, 0, stream>>>(prop, in_fea, out_fea,
                                                   W_r, b_r, W_z, b_z, W_t, b_t, n);
    }
}